// RoiBlock_7516192768624
// MI455X (gfx1250) — compile-verified
//
#include <hip/hip_runtime.h>

// ---------------------------------------------------------------------------
// RPN proposal kernel for MI455X (gfx1250), one workgroup per image.
//   B=16 images, N_AC=262144 anchors, top-6000 -> decode -> greedy NMS -> 1000
// Data-movement bound (~84 MB in -> ~4 us at 23.3 TB/s). No matmul shape, so
// no WMMA; gfx1250-specific paths exercised:
//   * global_load_async_to_lds_b128 + s_wait_asynccnt (ASYNCcnt), ping-pong
//     double-buffered so HBM tile fetch overlaps histogram/compaction work
//   * 128 KB LDS working set (staging x2, histograms, 8192-wide bitonic sort,
//     6000 x float4 box table) in a single workgroup per image
//   * wave32 __ballot suppression-bitmask NMS (each wave owns disjoint words)
// ---------------------------------------------------------------------------

#define NAC     262144
#define NSCORE  6000
#define NNMS    1000
#define TPB     1024          // 32 waves (wave32)
#define TILE    4096          // floats staged per tile (16 KB)
#define NTILES  (NAC / TILE)  // 64
#define SORTN   8192          // bitonic size (power of two >= NSCORE)
#define NMS_TH  0.7f

// LDS layout in 32-bit words (total 32768 words = 128 KB)
#define W_SCORE  0            // [0,8192)      sorted scores (dead after sort)
#define W_IDX    8192         // [8192,16384)  sorted anchor indices
#define W_STAGE0 16384        // [16384,20480) async staging buf 0 (phases A-C)
#define W_HIST   20480        // [20480,24576) histogram          (phases A-C)
#define W_STAGE1 24576        // [24576,28672) async staging buf 1 (phases A-C)
#define W_BOX    8192         // [8192,32192)  decoded boxes (overwrites above)
#define W_REM    32192        // [32192,32380) removed bitmask (188 words)
#define W_SCAL   32380        // scalars
#define S_CAB   0             // count strictly above coarse bin
#define S_CBIN  1             // coarse bin
#define S_TKEY  2             // 25-bit threshold key (bits >> 7)
#define S_G     3             // count strictly above threshold key
#define S_NEED  4             // ties needed from threshold key
#define S_MAIN  5             // compaction counter (strictly above)
#define S_TIE   6             // compaction counter (ties)

// gfx1250 async global->LDS copy: VDST vgpr = LDS byte offset, VADDR vgpr =
// 32-bit offset, SADDR sgpr pair = 64-bit base (GVS mode). Tracked by ASYNCcnt.
__device__ __forceinline__ void async_b128(unsigned lds_byte, unsigned g_byte,
                                           const float* sbase) {
  asm volatile("global_load_async_to_lds_b128 %0, %1, %2"
               :: "v"(lds_byte), "v"(g_byte), "s"(sbase) : "memory");
}
// Async loads complete in order: waiting "<= 1 outstanding" guarantees the
// older (previous tile) transfer has fully landed in LDS.
__device__ __forceinline__ void wait_async0() {
  asm volatile("s_wait_asynccnt 0" ::: "memory");
}
__device__ __forceinline__ void wait_async1() {
  asm volatile("s_wait_asynccnt 0x1" ::: "memory");
}

// Stream all NAC scores through the LDS ping-pong staging buffers; call
// body(bits, global_index) for each element this thread owns.
template <typename F>
__device__ __forceinline__ void stream_scores(const float* probB, int tid,
                                              unsigned* smem, F body) {
  // per-thread staging addresses, computed once (no memory-backed arrays)
  const unsigned ldsByte0 = W_STAGE0 * 4u + (unsigned)tid * 16u;
  const unsigned ldsByte1 = W_STAGE1 * 4u + (unsigned)tid * 16u;
  const unsigned wrd0     = W_STAGE0 + (unsigned)tid * 4u;
  const unsigned wrd1     = W_STAGE1 + (unsigned)tid * 4u;
  // prologue: issue tile 0 into buffer 0
  async_b128(ldsByte0, (unsigned)(tid * 4) * 4u, probB);
  for (int t = 0; t < NTILES; ++t) {
    const bool odd = (t & 1) != 0;
    if (t + 1 < NTILES) {
      // issue next tile into the other buffer (its previous consumer finished
      // at the end-of-iteration barrier of t-1), then wait for tile t only
      async_b128(odd ? ldsByte0 : ldsByte1,
                 (unsigned)((t + 1) * TILE + tid * 4) * 4u, probB);
      wait_async1();
    } else {
      wait_async0();
    }
    __syncthreads();   // tile t visible to all waves
    uint4 v = *reinterpret_cast<const uint4*>(smem + (odd ? wrd1 : wrd0));
    const int gi = t * TILE + tid * 4;
    body(v.x, gi + 0);
    body(v.y, gi + 1);
    body(v.z, gi + 2);
    body(v.w, gi + 3);
    __syncthreads();   // all reads of this buffer done before it is reissued
  }
}

__global__ __launch_bounds__(TPB)
void roi_topk_nms_kernel(const float* __restrict__ prob,
                         const float* __restrict__ del,
                         const float* __restrict__ anchors,
                         float* __restrict__ out) {
  __shared__ unsigned smem[32768];
  float* smf = reinterpret_cast<float*>(smem);

  const int tid = threadIdx.x;
  const int b   = blockIdx.x;
  const float* probB = prob + (size_t)b * NAC;

  // ---- init histogram + scalars -------------------------------------------
  for (int i = tid; i < 4096; i += TPB) smem[W_HIST + i] = 0u;
  if (tid < 16) smem[W_SCAL + tid] = 0u;
  __syncthreads();

  // ---- pass A: coarse 12-bit histogram of float bits (bits >> 19) ---------
  stream_scores(probB, tid, smem, [&](unsigned bits, int) {
    atomicAdd(&smem[W_HIST + (bits >> 19)], 1u);
  });
  if (tid == 0) {
    unsigned cum = 0, cab = 0, C = 0;
    for (int i = 4095; i >= 0; --i) {
      unsigned c = smem[W_HIST + i];
      if (cum + c >= NSCORE) { C = (unsigned)i; cab = cum; break; }
      cum += c;
    }
    smem[W_SCAL + S_CBIN] = C;
    smem[W_SCAL + S_CAB]  = cab;
  }
  __syncthreads();
  const unsigned Cbin = smem[W_SCAL + S_CBIN];
  const unsigned cab  = smem[W_SCAL + S_CAB];
  for (int i = tid; i < 4096; i += TPB) smem[W_HIST + i] = 0u;
  __syncthreads();

  // ---- pass B: fine 12-bit histogram within coarse bin ((bits>>7)&0xFFF) --
  stream_scores(probB, tid, smem, [&](unsigned bits, int) {
    if ((bits >> 19) == Cbin)
      atomicAdd(&smem[W_HIST + ((bits >> 7) & 0xFFFu)], 1u);
  });
  if (tid == 0) {
    unsigned cum = cab, G = cab, F = 0;
    for (int i = 4095; i >= 0; --i) {
      unsigned c = smem[W_HIST + i];
      if (cum + c >= NSCORE) { F = (unsigned)i; G = cum; break; }
      cum += c;
    }
    smem[W_SCAL + S_TKEY] = (Cbin << 12) | F;   // 25-bit key threshold
    smem[W_SCAL + S_G]    = G;
    smem[W_SCAL + S_NEED] = NSCORE - G;
  }
  __syncthreads();
  const unsigned tkey = smem[W_SCAL + S_TKEY];
  const unsigned G    = smem[W_SCAL + S_G];
  const unsigned need = smem[W_SCAL + S_NEED];

  // pad sort tail (slots [NSCORE, SORTN) sink to the end of a descending sort)
  for (int s = NSCORE + tid; s < SORTN; s += TPB) {
    smf[W_SCORE + s] = -1.0f;
    smem[W_IDX + s]  = 0u;
  }

  // ---- pass C: compaction of the top-6000 (score, index) into LDS ---------
  stream_scores(probB, tid, smem, [&](unsigned bits, int gi) {
    unsigned key = bits >> 7;
    if (key > tkey) {
      unsigned p = atomicAdd(&smem[W_SCAL + S_MAIN], 1u);
      smf[W_SCORE + p] = __uint_as_float(bits);
      smem[W_IDX + p]  = (unsigned)gi;
    } else if (key == tkey) {
      unsigned t2 = atomicAdd(&smem[W_SCAL + S_TIE], 1u);
      if (t2 < need) {
        unsigned p = G + t2;
        smf[W_SCORE + p] = __uint_as_float(bits);
        smem[W_IDX + p]  = (unsigned)gi;
      }
    }
  });

  // ---- bitonic sort, descending by score ----------------------------------
  for (unsigned k = 2; k <= SORTN; k <<= 1) {
    for (unsigned j = k >> 1; j > 0; j >>= 1) {
      __syncthreads();
      for (unsigned t = (unsigned)tid; t < SORTN; t += TPB) {
        unsigned ixj = t ^ j;
        if (ixj > t) {
          float a = smf[W_SCORE + t];
          float c = smf[W_SCORE + ixj];
          bool desc = ((t & k) == 0);
          if (desc ? (a < c) : (a > c)) {
            smf[W_SCORE + t]   = c;
            smf[W_SCORE + ixj] = a;
            unsigned ia = smem[W_IDX + t];
            smem[W_IDX + t]   = smem[W_IDX + ixj];
            smem[W_IDX + ixj] = ia;
          }
        }
      }
    }
  }
  __syncthreads();

  // ---- decode boxes (box table overwrites idx region; indices pre-read) ---
  unsigned myIdx[6];
  int cnt = 0;
  for (int s = tid; s < NSCORE; s += TPB) myIdx[cnt++] = smem[W_IDX + s];
  __syncthreads();
  cnt = 0;
  float4* bx = reinterpret_cast<float4*>(smf + W_BOX);
  for (int s = tid; s < NSCORE; s += TPB) {
    unsigned ai = myIdx[cnt++];
    float4 A = reinterpret_cast<const float4*>(anchors)[ai];
    float4 D = reinterpret_cast<const float4*>(del)[(size_t)b * NAC + ai];
    float ah = A.z - A.x, aw = A.w - A.y;
    float ay = A.x + 0.5f * ah, ax = A.y + 0.5f * aw;
    float cy = ay + D.x * ah,  cx = ax + D.y * aw;
    float h = ah * expf(D.z),  w = aw * expf(D.w);
    float y1 = fminf(fmaxf(cy - 0.5f * h, 0.0f), 1.0f);
    float x1 = fminf(fmaxf(cx - 0.5f * w, 0.0f), 1.0f);
    float y2 = fminf(fmaxf(cy + 0.5f * h, 0.0f), 1.0f);
    float x2 = fminf(fmaxf(cx + 0.5f * w, 0.0f), 1.0f);
    bx[s] = make_float4(y1, x1, y2, x2);
  }

  // ---- greedy NMS with wave32 ballot bitmask ------------------------------
  const int NWORDS = (NSCORE + 31) / 32;   // 188
  for (int i = tid; i < NWORDS; i += TPB) smem[W_REM + i] = 0u;
  __syncthreads();

  float4* outB = reinterpret_cast<float4*>(out + (size_t)b * (NNMS * 4));
  const int waveId = tid >> 5, lane = tid & 31;
  int kept = 0;
  for (int i = 0; i < NSCORE && kept < NNMS; ++i) {
    if (smem[W_REM + (i >> 5)] & (1u << (i & 31))) continue;   // uniform
    float4 bi = bx[i];
    if (tid == 0) outB[kept] = bi;                 // global_store_b128
    float area1 = (bi.z - bi.x) * (bi.w - bi.y);
    for (int wd = waveId; wd < NWORDS; wd += 32) { // wave owns word wd
      int j = wd * 32 + lane;
      bool sup = false;
      if (j < NSCORE) {
        float4 bj = bx[j];
        float yy1 = fmaxf(bi.x, bj.x), xx1 = fmaxf(bi.y, bj.y);
        float yy2 = fminf(bi.z, bj.z), xx2 = fminf(bi.w, bj.w);
        float inter = fmaxf(yy2 - yy1, 0.0f) * fmaxf(xx2 - xx1, 0.0f);
        float areas = (bj.z - bj.x) * (bj.w - bj.y);
        float iou = inter / (area1 + areas - inter + 1e-8f);
        sup = iou > NMS_TH;
      }
      unsigned m = (unsigned)__ballot(sup);
      if (lane == 0 && m) smem[W_REM + wd] |= m;
    }
    ++kept;
    __syncthreads();
  }
  __syncthreads();
  // zero-fill invalid tail rows (reference emits zeros when exhausted)
  float* outF = out + (size_t)b * (NNMS * 4);
  for (int r = kept * 4 + tid; r < NNMS * 4; r += TPB) outF[r] = 0.0f;
}

extern "C" void kernel_launch(void* const* d_in, const int* in_sizes, int n_in,
                              void* d_out, int out_size, void* d_ws, size_t ws_size,
                              hipStream_t stream) {
  const float* prob    = (const float*)d_in[0];   // [B, NAC, 1]
  const float* del     = (const float*)d_in[1];   // [B, NAC, 4]
  const float* anchors = (const float*)d_in[2];   // [NAC, 4]
  float* outp = (float*)d_out;                    // [B, NNMS, 4]
  int B = in_sizes[0] / NAC;
  roi_topk_nms_kernel<<<dim3(B), dim3(TPB), 0, stream>>>(prob, del, anchors, outp);
}